// CtIoULoss_64707977282025
// MI455X (gfx1250) — compile-verified
//
#include <hip/hip_runtime.h>
#include <stdint.h>

// ---------------- problem constants ----------------
#define Bn    8
#define Cn    80
#define Hn    128
#define Wn    128
#define HWn   (Hn*Wn)        // 16384
#define CHWn  (Cn*HWn)       // 1310720
#define TOT   (Bn*CHWn)      // 10485760
#define Kk    100
#define MOBJ  128
#define NGT   256
#define NBINS 1024
#define CANDMAX 4096
#define BETA  0.1f

// focal streaming config: TOT/4 float4 = 2,621,440 ; 1024 blk * 256 thr = 262,144 -> 10 iters exact
#define FOC_BLOCKS 1024
#define FOC_THREADS 256
#define FOC_STRIDE (FOC_BLOCKS*FOC_THREADS)
#define FOC_ITERS ((TOT/4)/FOC_STRIDE)

typedef __attribute__((ext_vector_type(2))) float v2f;
typedef __attribute__((ext_vector_type(8))) float v8f;

__device__ __forceinline__ float clampsig(float x) {
    float s = 1.0f / (1.0f + __expf(-x));
    return fminf(fmaxf(s, 1e-4f), 1.0f - 1e-4f);
}
__device__ __forceinline__ float clamp01(float x) { return fminf(fmaxf(x, 0.0f), 1.0f); }

// focal "negative" term: log(1-p) * p^2 * (1-gt)^4
__device__ __forceinline__ float negterm(float x, float gt) {
    float p = clampsig(x);
    float g = clamp01(gt);
    float q = 1.0f - g, q2 = q * q;
    return __logf(1.0f - p) * p * p * (q2 * q2);
}

// ---------------- CDNA5 async copy (global -> LDS), double-buffered ----------------
__device__ __forceinline__ unsigned lds_off(const void* p) {
    // generic shared pointer: addr[31:0] is the LDS byte offset (aperture scheme)
    return (unsigned)(uintptr_t)p;
}
__device__ __forceinline__ void async_ld128(const void* gp, unsigned lo) {
    asm volatile("global_load_async_to_lds_b128 %0, %1, off"
                 :: "v"(lo), "v"((unsigned long long)(uintptr_t)gp)
                 : "memory");
}
__device__ __forceinline__ void wait_async_le2() {
    asm volatile("s_wait_asynccnt 0x2" ::: "memory");
}
__device__ __forceinline__ void wait_async_0() {
    asm volatile("s_wait_asynccnt 0x0" ::: "memory");
}

// ---------------- wave32 sum-reduction via V_WMMA_F32_16X16X4_F32 ----------------
// A(16x4): lane L<16 -> (M=L, K=0/1), lane L>=16 -> (M=L-16, K=2/3). Put partial in a.x.
// B = ones(4x16) -> D[m,n] = rowsum[m] (replicated over n). Each lane sums its 8 D rows
// (rows 0-7 for lanes 0-15, rows 8-15 for lanes 16-31); shfl_xor(16) completes the total.
__device__ __forceinline__ float wave_reduce_wmma(float x) {
    v2f a; a.x = x; a.y = 0.0f;
    v2f b; b.x = 1.0f; b.y = 1.0f;
    v8f c = {};
    c = __builtin_amdgcn_wmma_f32_16x16x4_f32(false, a, false, b, (short)0, c, false, false);
    float s = c[0] + c[1] + c[2] + c[3] + c[4] + c[5] + c[6] + c[7];
    s += __shfl_xor(s, 16, 32);
    return s;
}

// ---------------- kernel 1: per-batch histogram of NMS-kept peak scores ----------------
__global__ void k_hist(const float* __restrict__ hm, unsigned* __restrict__ hist) {
    int e = blockIdx.x * 256 + threadIdx.x;
    if (e >= TOT) return;
    int b = e / CHWn, r = e % CHWn;
    int c = r / HWn, sind = r % HWn;
    int y = sind / Wn, x = sind % Wn;
    const float* base = hm + (size_t)(b * Cn + c) * HWn;
    float v = base[sind];
    float vmax = v;
    #pragma unroll
    for (int dy = -1; dy <= 1; ++dy)
        #pragma unroll
        for (int dx = -1; dx <= 1; ++dx) {
            int ny = y + dy, nx = x + dx;
            if (ny >= 0 && ny < Hn && nx >= 0 && nx < Wn)
                vmax = fmaxf(vmax, base[ny * Wn + nx]);
        }
    float s = clampsig(v);
    if (s == clampsig(vmax)) {  // sigmoid monotone: maxpool(sig)==sig(maxpool); clamp ties match ref
        int bin = (int)(s * 1024.0f);
        bin = min(max(bin, 0), NBINS - 1);
        atomicAdd(&hist[b * NBINS + bin], 1u);
    }
}

// ---------------- kernel 2: threshold bin so that count(bin >= t) >= K ----------------
__global__ void k_thresh(const unsigned* __restrict__ hist, int* __restrict__ tbin) {
    int b = threadIdx.x;
    if (b >= Bn) return;
    int cum = 0, t = 0;
    for (int bin = NBINS - 1; bin >= 0; --bin) {
        cum += (int)hist[b * NBINS + bin];
        if (cum >= Kk) { t = bin; break; }
    }
    tbin[b] = t;
}

// ---------------- kernel 3: collect candidates >= threshold bin ----------------
__global__ void k_collect(const float* __restrict__ hm, const int* __restrict__ tbin,
                          int* __restrict__ ccnt, float* __restrict__ cval, int* __restrict__ cidx) {
    int e = blockIdx.x * 256 + threadIdx.x;
    if (e >= TOT) return;
    int b = e / CHWn, r = e % CHWn;
    int c = r / HWn, sind = r % HWn;
    int y = sind / Wn, x = sind % Wn;
    const float* base = hm + (size_t)(b * Cn + c) * HWn;
    float v = base[sind];
    float vmax = v;
    #pragma unroll
    for (int dy = -1; dy <= 1; ++dy)
        #pragma unroll
        for (int dx = -1; dx <= 1; ++dx) {
            int ny = y + dy, nx = x + dx;
            if (ny >= 0 && ny < Hn && nx >= 0 && nx < Wn)
                vmax = fmaxf(vmax, base[ny * Wn + nx]);
        }
    float s = clampsig(v);
    if (s == clampsig(vmax)) {
        int bin = (int)(s * 1024.0f);
        bin = min(max(bin, 0), NBINS - 1);
        if (bin >= tbin[b]) {
            int p = atomicAdd(&ccnt[b], 1);
            if (p < CANDMAX) {
                cval[b * CANDMAX + p] = s;
                cidx[b * CANDMAX + p] = c * HWn + sind;  // flat C*H*W index (topk tie-break key)
            }
        }
    }
}

// ---------------- kernel 4: per-batch bitonic top-K, decode boxes, max-IoU vs GT ----------------
__global__ __launch_bounds__(1024) void k_topk(const float* __restrict__ cval, const int* __restrict__ cidx,
                                               const int* __restrict__ ccnt,
                                               const float* __restrict__ wh, const float* __restrict__ reg,
                                               const float* __restrict__ tbox, const int* __restrict__ tbidx,
                                               int* __restrict__ tki, float* __restrict__ tkv) {
    __shared__ float sv[CANDMAX];
    __shared__ int   si[CANDMAX];
    int b = blockIdx.x;
    int n = ccnt[b]; if (n > CANDMAX) n = CANDMAX;
    for (int i = threadIdx.x; i < CANDMAX; i += 1024) {
        if (i < n) { sv[i] = cval[b * CANDMAX + i]; si[i] = cidx[b * CANDMAX + i]; }
        else       { sv[i] = -1.0f;                 si[i] = 0x7fffffff; }
    }
    __syncthreads();
    // bitonic sort: value desc, index asc (matches jax top_k ordering)
    for (int k = 2; k <= CANDMAX; k <<= 1) {
        for (int j = k >> 1; j > 0; j >>= 1) {
            for (int i = threadIdx.x; i < CANDMAX; i += 1024) {
                int ixj = i ^ j;
                if (ixj > i) {
                    float va = sv[i], vb = sv[ixj];
                    int   ia = si[i], ib = si[ixj];
                    bool aBetter = (va > vb) || (va == vb && ia < ib);
                    bool desc = ((i & k) == 0);
                    if (desc ? !aBetter : aBetter) {
                        sv[i] = vb; si[i] = ib; sv[ixj] = va; si[ixj] = ia;
                    }
                }
            }
            __syncthreads();
        }
    }
    if (threadIdx.x < Kk) {
        int t = threadIdx.x;
        float val = sv[t];
        int   id  = si[t];
        if (val < 0.0f) { tki[b * Kk + t] = -1; tkv[b * Kk + t] = 0.0f; return; }
        int sind = id % HWn;
        int y = sind / Wn, x = sind % Wn;
        float rx = reg[(size_t)(b * 2 + 0) * HWn + sind];
        float ry = reg[(size_t)(b * 2 + 1) * HWn + sind];
        float w0 = wh [(size_t)(b * 2 + 0) * HWn + sind];
        float w1 = wh [(size_t)(b * 2 + 1) * HWn + sind];
        float xs = (float)x + rx, ys = (float)y + ry;
        float bx1 = xs - 0.5f * w0, by1 = ys - 0.5f * w1;
        float bx2 = xs + 0.5f * w0, by2 = ys + 0.5f * w1;
        float a1 = fmaxf(bx2 - bx1, 0.0f) * fmaxf(by2 - by1, 0.0f);
        float best = 0.0f;
        for (int g = 0; g < NGT; ++g) {
            if (tbidx[g] != b) continue;
            float gx1 = tbox[g*4+0], gy1 = tbox[g*4+1], gx2 = tbox[g*4+2], gy2 = tbox[g*4+3];
            float a2 = fmaxf(gx2 - gx1, 0.0f) * fmaxf(gy2 - gy1, 0.0f);
            float iw = fmaxf(fminf(bx2, gx2) - fmaxf(bx1, gx1), 0.0f);
            float ih = fmaxf(fminf(by2, gy2) - fmaxf(by1, gy1), 0.0f);
            float inter = iw * ih;
            best = fmaxf(best, inter / fmaxf(a1 + a2 - inter, 1e-7f));
        }
        tki[b * Kk + t] = id;
        tkv[b * Kk + t] = best;
    }
}

// ---------------- kernel 5: focal loss main stream (iou_map == 0 everywhere) ----------------
// Double-buffered async global->LDS pipeline; WMMA wave reduction; f64 atomic per wave.
__global__ __launch_bounds__(256) void k_focal(const float* __restrict__ hm,
                                               const float* __restrict__ hmt,
                                               double* __restrict__ acc) {
    __shared__ float4 sh_hm[2][FOC_THREADS];
    __shared__ float4 sh_tg[2][FOC_THREADS];
    const float4* hm4 = (const float4*)hm;
    const float4* tg4 = (const float4*)hmt;
    int tid = blockIdx.x * FOC_THREADS + threadIdx.x;

    async_ld128(hm4 + tid, lds_off(&sh_hm[0][threadIdx.x]));
    async_ld128(tg4 + tid, lds_off(&sh_tg[0][threadIdx.x]));

    float accv = 0.0f;
    for (int it = 0; it < FOC_ITERS; ++it) {
        int buf = it & 1;
        if (it + 1 < FOC_ITERS) {
            int g = tid + (it + 1) * FOC_STRIDE;
            async_ld128(hm4 + g, lds_off(&sh_hm[buf ^ 1][threadIdx.x]));
            async_ld128(tg4 + g, lds_off(&sh_tg[buf ^ 1][threadIdx.x]));
            wait_async_le2();   // 2 newest outstanding -> current buffer landed (in-order loads)
        } else {
            wait_async_0();
        }
        float4 hv = sh_hm[buf][threadIdx.x];
        float4 tv = sh_tg[buf][threadIdx.x];
        accv += negterm(hv.x, tv.x) + negterm(hv.y, tv.y)
              + negterm(hv.z, tv.z) + negterm(hv.w, tv.w);
    }
    float wsum = wave_reduce_wmma(accv);   // EXEC all-ones here (uniform flow, full blocks)
    if ((threadIdx.x & 31) == 0) atomicAdd(&acc[0], (double)wsum);
}

// ---------------- kernel 6: sparse focal corrections at the 800 scattered IoU points ----------------
__global__ void k_corr(const float* __restrict__ hm, const float* __restrict__ hmt,
                       const int* __restrict__ tki, const float* __restrict__ tkv,
                       double* __restrict__ acc) {
    int t = blockIdx.x * blockDim.x + threadIdx.x;
    if (t >= Bn * Kk) return;
    int id = tki[t];
    if (id < 0) return;
    int b = t / Kk;
    size_t off = (size_t)b * CHWn + id;
    float p  = clampsig(hm[off]);
    float g0 = clamp01(hmt[off]);
    float g1 = clamp01(hmt[off] + BETA * tkv[t]);
    float q0 = 1.0f - g0, q02 = q0 * q0;
    float oldneg = __logf(1.0f - p) * p * p * (q02 * q02);
    if (g1 >= 1.0f) {
        float pos = __logf(p) * (1.0f - p) * (1.0f - p);
        atomicAdd(&acc[1], (double)pos);
        atomicAdd(&acc[2], 1.0);
        atomicAdd(&acc[0], -(double)oldneg);
    } else {
        float q1 = 1.0f - g1, q12 = q1 * q1;
        float newneg = __logf(1.0f - p) * p * p * (q12 * q12);
        atomicAdd(&acc[0], (double)(newneg - oldneg));
    }
}

// ---------------- kernel 7: masked L1 losses (wh + offset), one block ----------------
__global__ __launch_bounds__(1024) void k_reg(const float* __restrict__ wh, const float* __restrict__ reg,
                                              const float* __restrict__ wht, const float* __restrict__ regt,
                                              const int* __restrict__ mask, const int* __restrict__ ind,
                                              double* __restrict__ acc) {
    __shared__ float s0[1024], s1[1024], s2[1024];
    int t = threadIdx.x;           // B*MOBJ == 1024 exactly
    int b = t / MOBJ;
    float mk = (float)mask[t];
    int iv = ind[t];
    float whs = 0.0f, offs = 0.0f;
    #pragma unroll
    for (int ch = 0; ch < 2; ++ch) {
        float pw = wh [(size_t)(b * 2 + ch) * HWn + iv];
        float pr = reg[(size_t)(b * 2 + ch) * HWn + iv];
        whs  += fabsf(pw * mk - wht [t * 2 + ch] * mk);
        offs += fabsf(pr * mk - regt[t * 2 + ch] * mk);
    }
    s0[t] = whs; s1[t] = offs; s2[t] = 2.0f * mk;
    __syncthreads();
    for (int s = 512; s > 0; s >>= 1) {
        if (t < s) { s0[t] += s0[t + s]; s1[t] += s1[t + s]; s2[t] += s2[t + s]; }
        __syncthreads();
    }
    if (t == 0) { acc[3] = (double)s0[0]; acc[4] = (double)s2[0]; acc[5] = (double)s1[0]; }
}

// ---------------- kernel 8: combine ----------------
__global__ void k_fin(const double* __restrict__ acc, float* __restrict__ out) {
    if (threadIdx.x == 0 && blockIdx.x == 0) {
        double neg = acc[0], pos = acc[1], np = acc[2];
        double hm_loss  = (np > 0.0) ? (-(pos + neg) / fmax(np, 1.0)) : (-neg);
        double wh_loss  = acc[3] / (acc[4] + 1e-4);
        double off_loss = acc[5] / (acc[4] + 1e-4);
        double loss = 1.0 * hm_loss + 0.1 * wh_loss + 1.0 * off_loss;
        out[0] = (float)loss; out[1] = (float)hm_loss;
        out[2] = (float)wh_loss; out[3] = (float)off_loss;
    }
}

// ---------------- launch ----------------
extern "C" void kernel_launch(void* const* d_in, const int* in_sizes, int n_in,
                              void* d_out, int out_size, void* d_ws, size_t ws_size,
                              hipStream_t stream) {
    (void)in_sizes; (void)n_in; (void)out_size; (void)ws_size;
    const float* hm   = (const float*)d_in[0];
    const float* wh   = (const float*)d_in[1];
    const float* reg  = (const float*)d_in[2];
    const float* hmt  = (const float*)d_in[3];
    const float* wht  = (const float*)d_in[4];
    const float* regt = (const float*)d_in[5];
    const int*   rmsk = (const int*)d_in[6];
    const int*   ind  = (const int*)d_in[7];
    const float* tbox = (const float*)d_in[8];
    const int*   tbid = (const int*)d_in[9];
    float* out = (float*)d_out;

    char* ws = (char*)d_ws;
    double*   acc  = (double*)ws;                                 // [6] accumulators
    unsigned* hist = (unsigned*)(ws + 64);                        // B*1024 bins
    int*      tbin = (int*)(ws + 64 + Bn * NBINS * 4);            // @32832
    int*      ccnt = tbin + Bn;                                   // @32864
    float*    cval = (float*)(ws + 32896);                        // B*4096 f32
    int*      cidx = (int*)(ws + 32896 + Bn * CANDMAX * 4);       // B*4096 i32
    int*      tki  = (int*)(ws + 32896 + 2 * Bn * CANDMAX * 4);   // B*K i32
    float*    tkv  = (float*)(ws + 32896 + 2 * Bn * CANDMAX * 4 + Bn * Kk * 4); // B*K f32

    // zero accumulators, histograms, candidate counters (graph-capture-safe memset node)
    hipMemsetAsync(d_ws, 0, 33024, stream);

    k_hist   <<<(TOT + 255) / 256, 256, 0, stream>>>(hm, hist);
    k_thresh <<<1, 32, 0, stream>>>(hist, tbin);
    k_collect<<<(TOT + 255) / 256, 256, 0, stream>>>(hm, tbin, ccnt, cval, cidx);
    k_topk   <<<Bn, 1024, 0, stream>>>(cval, cidx, ccnt, wh, reg, tbox, tbid, tki, tkv);
    k_focal  <<<FOC_BLOCKS, FOC_THREADS, 0, stream>>>(hm, hmt, acc);
    k_corr   <<<(Bn * Kk + 255) / 256, 256, 0, stream>>>(hm, hmt, tki, tkv, acc);
    k_reg    <<<1, 1024, 0, stream>>>(wh, reg, wht, regt, rmsk, ind, acc);
    k_fin    <<<1, 64, 0, stream>>>(acc, out);
}